// FilterDetections_6992206758510
// MI455X (gfx1250) — compile-verified
//
#include <hip/hip_runtime.h>
#include <hip/hip_bf16.h>
#include <stdint.h>

#define BB 4
#define NN 100000
#define CC 16
#define NBINS 8192
#define CAP 4096
#define MAXDET 300
#define THRESH 0.5f
#define TILE 4000            // floats per TDM tile (16000 B); 400 tiles per image
#define TILES_PER_BLOCK 8
#define BLOCKS_PER_IMG 50    // 50 * 8 * 4000 = 1,600,000 = NN*CC exactly

typedef unsigned int u32x4 __attribute__((ext_vector_type(4)));
typedef int          i32x8 __attribute__((ext_vector_type(8)));
typedef int          i32x4 __attribute__((ext_vector_type(4)));

// ---------------------------------------------------------------------------
// Kernel 0: zero the integer workspace (histograms + cutoff + candidate count)
// ---------------------------------------------------------------------------
__global__ void fd_init_kernel(int* __restrict__ p, int n) {
    int i = blockIdx.x * blockDim.x + threadIdx.x;
    int stride = gridDim.x * blockDim.x;
    for (; i < n; i += stride) p[i] = 0;
}

// ---------------------------------------------------------------------------
// Kernel 1: per-image score histogram. TDM (tensor_load_to_lds) stages 16 KB
// tiles into LDS, double-buffered; waves histogram out of LDS with ds atomics.
// ---------------------------------------------------------------------------
__global__ void fd_hist_kernel(const float* __restrict__ cls,
                               int* __restrict__ hist) {
    __shared__ float buf[2][TILE];
    __shared__ int shist[NBINS];

    const int b = blockIdx.y;
    const float* base = cls + (size_t)b * NN * CC;

    for (int j = threadIdx.x; j < NBINS; j += blockDim.x) shist[j] = 0;

    auto issue = [&](int t, int p) {
        const float* src =
            base + (size_t)(blockIdx.x * TILES_PER_BLOCK + t) * TILE;
        unsigned long long ga = (unsigned long long)(const void*)src;
        // Generic LDS pointers carry the LDS byte offset in addr[31:0].
        unsigned lds = (unsigned)(unsigned long long)(void*)&buf[p][0];

        // D# group 0: count=1 | lds_addr | global_addr[56:0] | type=2
        u32x4 g0 = { 1u,
                     lds,
                     (unsigned)(ga & 0xFFFFFFFFu),
                     (unsigned)((ga >> 32) & 0x01FFFFFFu) | (2u << 30) };
        // D# group 1: data_size=4B(code 2); 1-D tile: tensor_dim0 = tile_dim0
        // = TILE elements, tensor_dim1 = tile_dim1 = 1, stride0 = TILE.
        i32x8 g1 = { (int)(2u << 16),                       // data_size=4B
                     (int)((TILE & 0xFFFF) << 16),          // tensor_dim0 lo16
                     (int)((TILE >> 16) | (1u << 16)),      // dim0 hi16 | dim1 lo16=1
                     (int)((TILE & 0xFFFF) << 16),          // dim1 hi=0 | tile_dim0
                     1,                                     // tile_dim1=1, tile_dim2=0
                     TILE,                                  // tensor_dim0_stride lo32
                     0, 0 };
        i32x4 z4 = { 0, 0, 0, 0 };
        i32x8 z8 = { 0, 0, 0, 0, 0, 0, 0, 0 };
        __builtin_amdgcn_tensor_load_to_lds(g0, g1, z4, z4, z8, 0);
    };

    if (threadIdx.x == 0) issue(0, 0);

    for (int t = 0; t < TILES_PER_BLOCK; ++t) {
        __builtin_amdgcn_s_wait_tensorcnt(0);   // issuing wave: tile t landed
        __syncthreads();                        // publish LDS tile to all waves
        if (threadIdx.x == 0 && t + 1 < TILES_PER_BLOCK)
            issue(t + 1, (t + 1) & 1);          // prefetch next tile (async)
        const float* cur = buf[t & 1];
        for (int j = threadIdx.x; j < TILE; j += blockDim.x) {
            float s = cur[j];
            if (s > THRESH) {
                int bin = (int)((s - THRESH) * (float)(NBINS) / THRESH);
                bin = bin < 0 ? 0 : (bin >= NBINS ? NBINS - 1 : bin);
                atomicAdd(&shist[bin], 1);      // ds_add_u32
            }
        }
        __syncthreads();                        // done reading before re-fill
    }

    int* gh = hist + b * NBINS;
    for (int j = threadIdx.x; j < NBINS; j += blockDim.x) {
        int v = shist[j];
        if (v) atomicAdd(&gh[j], v);
    }
}

// ---------------------------------------------------------------------------
// Kernel 2: find the cutoff bin: highest k with suffix-count(k) >= MAXDET.
// ---------------------------------------------------------------------------
__global__ void fd_cutoff_kernel(const int* __restrict__ hist,
                                 int* __restrict__ cutoff) {
    __shared__ int scan[256];
    const int b = blockIdx.x;
    const int* h = hist + b * NBINS;
    const int PER = NBINS / 256;                // 32 bins per thread
    const int t = threadIdx.x;

    int sum = 0;
    for (int j = 0; j < PER; ++j)
        sum += h[NBINS - 1 - (t * PER + j)];    // high bins first

    scan[t] = sum;
    __syncthreads();
    for (int off = 1; off < 256; off <<= 1) {   // inclusive Hillis-Steele
        int add = (t >= off) ? scan[t - off] : 0;
        __syncthreads();
        scan[t] += add;
        __syncthreads();
    }
    int run = scan[t] - sum;                    // exclusive prefix (from top)
    for (int j = 0; j < PER; ++j) {
        int bin = NBINS - 1 - (t * PER + j);
        int c = h[bin];
        if (run < MAXDET && run + c >= MAXDET) cutoff[b] = bin;  // unique
        run += c;
    }
    // if total valid < MAXDET, cutoff stays 0 (set by init kernel)
}

// ---------------------------------------------------------------------------
// Kernel 3: collect candidate keys (scoreBits<<32 | ~classMajorIdx).
// ---------------------------------------------------------------------------
__global__ void fd_collect_kernel(const float* __restrict__ cls,
                                  const int* __restrict__ cutoff,
                                  int* __restrict__ candCount,
                                  unsigned long long* __restrict__ cand) {
    const int b = blockIdx.y;
    const int cut = cutoff[b];
    const float* base = cls + (size_t)b * NN * CC;
    const int total = NN * CC;
    int e = (blockIdx.x * blockDim.x + threadIdx.x) * 4;
    const int stride = gridDim.x * blockDim.x * 4;
    for (; e < total; e += stride) {
        float4 v = *(const float4*)(base + e);
        float sv[4] = { v.x, v.y, v.z, v.w };
#pragma unroll
        for (int k = 0; k < 4; ++k) {
            float s = sv[k];
            if (s > THRESH) {
                int bin = (int)((s - THRESH) * (float)(NBINS) / THRESH);
                bin = bin >= NBINS ? NBINS - 1 : bin;
                if (bin >= cut) {
                    int idx = e + k;                 // (anchor,class) linear
                    int n = idx / CC;
                    int c = idx - n * CC;
                    unsigned fidx = (unsigned)(c * NN + n);   // class-major
                    unsigned long long key =
                        ((unsigned long long)__float_as_uint(s) << 32) |
                        (unsigned long long)(0xFFFFFFFFu - fidx);
                    int pos = atomicAdd(&candCount[b], 1);
                    if (pos < CAP) cand[(size_t)b * CAP + pos] = key;
                }
            }
        }
    }
}

// ---------------------------------------------------------------------------
// Kernel 4: one block per image — bitonic sort (desc) in LDS, emit top-300.
// ---------------------------------------------------------------------------
__global__ void fd_topk_kernel(const unsigned long long* __restrict__ cand,
                               const int* __restrict__ candCount,
                               const float* __restrict__ poses,
                               float* __restrict__ out) {
    __shared__ unsigned long long keys[CAP];    // 32 KB
    const int b = blockIdx.x;
    int cnt = candCount[b];
    if (cnt > CAP) cnt = CAP;

    for (int i = threadIdx.x; i < CAP; i += blockDim.x)
        keys[i] = (i < cnt) ? cand[(size_t)b * CAP + i] : 0ull;
    __syncthreads();

    for (int k = 2; k <= CAP; k <<= 1) {
        for (int j = k >> 1; j > 0; j >>= 1) {
            for (int i = threadIdx.x; i < CAP; i += blockDim.x) {
                int p = i ^ j;
                if (p > i) {
                    unsigned long long a = keys[i], c = keys[p];
                    bool desc = ((i & k) == 0);
                    if ((a < c) == desc) { keys[i] = c; keys[p] = a; }
                }
            }
            __syncthreads();
        }
    }

    float* out_scores  = out;                                   // [B][300]
    float* out_labels  = out + BB * MAXDET;                     // [B][300]
    float* out_poses   = out + 2 * BB * MAXDET;                 // [B][300][12]
    float* out_indices = out + 2 * BB * MAXDET + BB * MAXDET * 12;

    for (int i = threadIdx.x; i < MAXDET; i += blockDim.x) {
        unsigned long long key = keys[i];
        unsigned sb = (unsigned)(key >> 32);
        float* po = out_poses + ((size_t)b * MAXDET + i) * 12;
        if (sb != 0u) {
            float s = __uint_as_float(sb);
            unsigned fidx = 0xFFFFFFFFu - (unsigned)(key & 0xFFFFFFFFu);
            int label  = (int)(fidx / NN);
            int anchor = (int)(fidx - (unsigned)label * NN);
            out_scores [b * MAXDET + i] = s;
            out_labels [b * MAXDET + i] = (float)label;
            out_indices[b * MAXDET + i] = (float)anchor;
            const float* pp =
                poses + (((size_t)b * NN + anchor) * CC + label) * 12;
#pragma unroll
            for (int q = 0; q < 12; ++q) po[q] = pp[q];
        } else {
            out_scores [b * MAXDET + i] = -1.0f;
            out_labels [b * MAXDET + i] = -1.0f;
            out_indices[b * MAXDET + i] = -1.0f;
#pragma unroll
            for (int q = 0; q < 12; ++q) po[q] = -1.0f;
        }
    }
}

// ---------------------------------------------------------------------------
extern "C" void kernel_launch(void* const* d_in, const int* in_sizes, int n_in,
                              void* d_out, int out_size, void* d_ws, size_t ws_size,
                              hipStream_t stream) {
    // inputs (setup_inputs order): boxes3D, classification, poses, confidence
    const float* cls   = (const float*)d_in[1];
    const float* poses = (const float*)d_in[2];

    // workspace layout
    int* hist      = (int*)d_ws;                    // B*NBINS ints
    int* cutoff    = hist + BB * NBINS;             // B ints
    int* candCount = cutoff + BB;                   // B ints
    size_t candOff = (size_t)(BB * NBINS + 2 * BB) * sizeof(int); // 8B aligned
    unsigned long long* cand = (unsigned long long*)((char*)d_ws + candOff);

    fd_init_kernel<<<64, 256, 0, stream>>>(hist, BB * NBINS + 2 * BB);
    fd_hist_kernel<<<dim3(BLOCKS_PER_IMG, BB), 256, 0, stream>>>(cls, hist);
    fd_cutoff_kernel<<<BB, 256, 0, stream>>>(hist, cutoff);
    fd_collect_kernel<<<dim3(200, BB), 256, 0, stream>>>(cls, cutoff,
                                                         candCount, cand);
    fd_topk_kernel<<<BB, 512, 0, stream>>>(cand, candCount, poses,
                                           (float*)d_out);
}